// CORLoss_28741921145431
// MI455X (gfx1250) — compile-verified
//
#include <hip/hip_runtime.h>

typedef float v2f __attribute__((ext_vector_type(2)));
typedef float v4f __attribute__((ext_vector_type(4)));
typedef float v8f __attribute__((ext_vector_type(8)));

#define ROWS_TOTAL       131072
#define NCOL             128
#define EPSV             0.001f
#define WAVES_PER_BLOCK  2
#define TILEROWS         16
#define ROWSTRIDE        132                   // padded floats/row in LDS (bank-conflict-free, 16B-aligned rows)
#define ROWBYTES         (ROWSTRIDE * 4)       // 528
#define TILEF            (TILEROWS * ROWSTRIDE)
#define NBLOCKS          (ROWS_TOTAL / (TILEROWS * WAVES_PER_BLOCK))  // 4096

// Accumulate row sums of A-tile into 16x16 f32 accumulator via FP32 WMMA with all-ones B.
#define WMMA_ACC(acc, amat)                                                     \
  acc = __builtin_amdgcn_wmma_f32_16x16x4_f32(false, (amat), false, ones,       \
                                              (short)0, (acc), false, false)

__global__ __launch_bounds__(WAVES_PER_BLOCK * 32)
void corloss_main(const float* __restrict__ d1, const float* __restrict__ d2,
                  float* __restrict__ partial) {
  __shared__ float lds[WAVES_PER_BLOCK * 2 * TILEF];          // staged tiles (d1 then d2 per wave)
  __shared__ float stats[WAVES_PER_BLOCK * TILEROWS * 8];     // per-row scalar stats
  __shared__ float wsum[WAVES_PER_BLOCK];

  const int lane = threadIdx.x & 31;
  const int wave = threadIdx.x >> 5;
  const int tile = blockIdx.x * WAVES_PER_BLOCK + wave;
  const size_t row0 = (size_t)tile * TILEROWS;

  float* l1 = &lds[wave * 2 * TILEF];
  float* l2 = l1 + TILEF;

  // ---------------- Phase 0: async DMA global -> LDS (one 512B row per instruction) --------------
  {
    const unsigned l1off = (unsigned)(size_t)(void*)l1;
    const unsigned l2off = (unsigned)(size_t)(void*)l2;
    const unsigned long long g1 = (unsigned long long)(const void*)(d1 + row0 * NCOL);
    const unsigned long long g2 = (unsigned long long)(const void*)(d2 + row0 * NCOL);
#pragma unroll
    for (int r = 0; r < TILEROWS; ++r) {
      unsigned dst1 = l1off + (unsigned)(r * ROWBYTES) + (unsigned)(lane * 16);
      unsigned long long src1 = g1 + (unsigned long long)(r * NCOL * 4 + lane * 16);
      asm volatile("global_load_async_to_lds_b128 %0, %1, off"
                   :: "v"(dst1), "v"(src1) : "memory");
      unsigned dst2 = l2off + (unsigned)(r * ROWBYTES) + (unsigned)(lane * 16);
      unsigned long long src2 = g2 + (unsigned long long)(r * NCOL * 4 + lane * 16);
      asm volatile("global_load_async_to_lds_b128 %0, %1, off"
                   :: "v"(dst2), "v"(src2) : "memory");
    }
    asm volatile("s_wait_asynccnt 0" ::: "memory");
  }

  const int m = lane & 15;   // row within tile owned by this lane
  const int kh = lane >> 4;  // A-layout K-half / row-half selector

  // ---------------- Phase 1: five row-sum reductions via FP32 WMMA (B = ones) -------------------
  v8f accS1 = {0,0,0,0,0,0,0,0}, accS2 = accS1, accQ1 = accS1, accQ2 = accS1, accP = accS1;
  {
    v2f ones; ones.x = 1.0f; ones.y = 1.0f;
    const float* r1 = l1 + m * ROWSTRIDE + kh * 2;
    const float* r2 = l2 + m * ROWSTRIDE + kh * 2;
#pragma unroll
    for (int kc = 0; kc < 32; ++kc) {
      v2f a1 = *(const v2f*)(r1 + kc * 4);   // A[m][4kc + 2kh + {0,1}]
      v2f a2 = *(const v2f*)(r2 + kc * 4);
      v2f q1 = a1 * a1;
      v2f q2 = a2 * a2;
      v2f pp = a1 * a2;
      WMMA_ACC(accS1, a1);
      WMMA_ACC(accS2, a2);
      WMMA_ACC(accQ1, q1);
      WMMA_ACC(accQ2, q2);
      WMMA_ACC(accP,  pp);
    }
  }

  // D layout: lanes 0-15 hold rows 0-7 in v[0..7] (n=lane), lanes 16-31 hold rows 8-15.
  if ((lane & 15) == 0) {  // lanes 0 and 16 publish their 8 rows
    const int rb = (lane >> 4) * 8;
    float* st = &stats[(wave * TILEROWS + rb) * 8];
#pragma unroll
    for (int j = 0; j < 8; ++j) {
      st[j * 8 + 0] = accS1[j];
      st[j * 8 + 1] = accS2[j];
      st[j * 8 + 2] = accQ1[j];
      st[j * 8 + 3] = accQ2[j];
      st[j * 8 + 4] = accP[j];
    }
  }
  __syncthreads();

  // ---------------- Per-row correlation terms ----------------------------------------------------
  float aab, tl1;
  {
    const float* st = &stats[(wave * TILEROWS + m) * 8];
    float s1 = st[0], s2 = st[1], q1 = st[2], q2 = st[3], p12 = st[4];
    const float invN = 1.0f / (float)NCOL;
    const float invN1 = 1.0f / (float)(NCOL - 1);
    float m1 = s1 * invN, m2 = s2 * invN;
    float m1e = m1 + EPSV, m2e = m2 + EPSV;           // reference adds EPS to means before cov
    float cov = (p12 - m2e * s1 - m1e * s2 + (float)NCOL * m1e * m2e) * invN1;
    float var1 = fmaxf((q1 - s1 * m1) * invN1, 0.0f);
    float var2 = fmaxf((q2 - s2 * m2) * invN1, 0.0f);
    float cor = cov / (sqrtf(var1) * sqrtf(var2) + EPSV);
    float c3 = cor * cor * cor;
    tl1 = -logf((c3 + 1.0f + EPSV) * 0.5f);
    aab = fabsf(c3);
  }

  // ---------------- Phase 2: softmax L1 term. Lane pair (L, L+16) splits row m in halves. --------
  float tl2;
  {
    float* e1 = l1 + m * ROWSTRIDE + kh * 64;
    float* e2 = l2 + m * ROWSTRIDE + kh * 64;
    float se1 = 0.0f, se2 = 0.0f;
#pragma unroll
    for (int j = 0; j < 16; ++j) {
      v4f x1 = *(v4f*)(e1 + j * 4);
      v4f x2 = *(v4f*)(e2 + j * 4);
      v4f t1, t2;
#pragma unroll
      for (int k = 0; k < 4; ++k) { t1[k] = expf(x1[k]); t2[k] = expf(x2[k]); }
      *(v4f*)(e1 + j * 4) = t1;   // cache exp in-place, avoid recompute in pass 2b
      *(v4f*)(e2 + j * 4) = t2;
      se1 += t1[0] + t1[1] + t1[2] + t1[3];
      se2 += t2[0] + t2[1] + t2[2] + t2[3];
    }
    se1 += __shfl_xor(se1, 16, 32);
    se2 += __shfl_xor(se2, 16, 32);
    const float inv1 = 1.0f / se1, inv2 = 1.0f / se2;
    float sabs = 0.0f;
#pragma unroll
    for (int j = 0; j < 16; ++j) {
      v4f t1 = *(v4f*)(e1 + j * 4);
      v4f t2 = *(v4f*)(e2 + j * 4);
#pragma unroll
      for (int k = 0; k < 4; ++k) sabs += fabsf(t1[k] * inv1 - t2[k] * inv2);
    }
    sabs += __shfl_xor(sabs, 16, 32);
    tl2 = sabs * (1.0f / (float)NCOL);
  }

  // ---------------- Reduce: per-wave -> per-block partial (deterministic) ------------------------
  float loss = aab * tl1 + (1.0f - aab) * tl2;
  float v = (lane < 16) ? loss : 0.0f;  // lanes 16-31 are duplicates of rows 0-15
#pragma unroll
  for (int s = 16; s; s >>= 1) v += __shfl_xor(v, s, 32);
  if (lane == 0) wsum[wave] = v;
  __syncthreads();
  if (threadIdx.x == 0) partial[blockIdx.x] = wsum[0] + wsum[1];
}

__global__ __launch_bounds__(256)
void corloss_reduce(const float* __restrict__ partial, float* __restrict__ out) {
  __shared__ float sm[256];
  float s = 0.0f;
  for (int i = threadIdx.x; i < NBLOCKS; i += 256) s += partial[i];
  sm[threadIdx.x] = s;
  __syncthreads();
  for (int stride = 128; stride; stride >>= 1) {
    if ((int)threadIdx.x < stride) sm[threadIdx.x] += sm[threadIdx.x + stride];
    __syncthreads();
  }
  if (threadIdx.x == 0) out[0] = sm[0];
}

extern "C" void kernel_launch(void* const* d_in, const int* in_sizes, int n_in,
                              void* d_out, int out_size, void* d_ws, size_t ws_size,
                              hipStream_t stream) {
  const float* d1 = (const float*)d_in[0];
  const float* d2 = (const float*)d_in[1];
  float* partial = (float*)d_ws;        // NBLOCKS floats = 16 KB
  float* out = (float*)d_out;
  corloss_main<<<NBLOCKS, WAVES_PER_BLOCK * 32, 0, stream>>>(d1, d2, partial);
  corloss_reduce<<<1, 256, 0, stream>>>(partial, out);
}